// SimpleSAGE_65584150609961
// MI455X (gfx1250) — compile-verified
//
#include <hip/hip_runtime.h>

typedef __attribute__((ext_vector_type(2))) float v2f;
typedef __attribute__((ext_vector_type(8))) float v8f;

#define N_NODES 100000
#define D_IN    128
#define D_OUT   64
#define N_TILES (N_NODES / 16)   // 6250, exact
#define NPAIRS  (D_IN / 2 * D_OUT)  // 4096 float2 pairs per weight matrix (32 KB)

// ---------------------------------------------------------------------------
// Kernel 1: zero the scratch (agg sums + counts). Must run every call since
// the scatter kernel accumulates and graph replays don't re-poison d_ws.
// ---------------------------------------------------------------------------
__global__ void sage_zero_ws(float* __restrict__ ws, int total) {
    int i = blockIdx.x * blockDim.x + threadIdx.x;
    if (i < total) ws[i] = 0.0f;
}

// ---------------------------------------------------------------------------
// Kernel 2: edge scatter. One wave32 per edge; each lane handles 4 floats of
// the 128-wide feature row. Non-returning f32 atomics -> global_atomic_add_f32.
// Lane 0 bumps the in-degree count. x (51 MB) lives in the 192 MB L2, so the
// random gathers mostly hit cache; the atomics are the HBM/L2 floor.
// ---------------------------------------------------------------------------
__global__ void sage_edge_scatter(const float* __restrict__ x,
                                  const int*   __restrict__ src,
                                  const int*   __restrict__ dst,
                                  float*       __restrict__ agg,
                                  float*       __restrict__ cnt,
                                  int E) {
    long long t    = (long long)blockIdx.x * blockDim.x + threadIdx.x;
    long long e    = t >> 5;
    int       lane = (int)(t & 31);
    if (e >= E) return;
    int s = src[e];
    int d = dst[e];
    const float4 v = *(const float4*)(x + (size_t)s * D_IN + lane * 4);
    float* ad = agg + (size_t)d * D_IN + lane * 4;
    atomicAdd(ad + 0, v.x);
    atomicAdd(ad + 1, v.y);
    atomicAdd(ad + 2, v.z);
    atomicAdd(ad + 3, v.w);
    if (lane == 0) atomicAdd(cnt + d, 1.0f);
}

// ---------------------------------------------------------------------------
// Kernel 3: out = relu( (agg/max(cnt,1)) @ W_l + x @ W_r + b )
// 256 threads = 8 waves; each wave owns a 16-row node tile and produces the
// full 16x64 output block with V_WMMA_F32_16X16X4_F32 (fp32-exact).
//
// W_l/W_r are staged into LDS *pre-swizzled into B-fragment order*: the pair
// (W[kk][col], W[kk+1][col]) a lane needs is stored contiguously, so each
// B fragment is ONE aligned ds_load_b64 straight into an even VGPR pair —
// no ds_load_2addr + v_mov repacking (which dominated the round-1 asm).
//
// pairIdx = (k0/4)*128 + half*64 + col   (col = n*16 + lid, half = lane>>4)
//
// fp32 WMMA fragment layouts (ISA 7.12.2):
//   A (16x4): lane<16 -> (M=lane, K=k0,k0+1), lane>=16 -> (M=lane-16, K=k0+2,k0+3)
//   B (4x16): same with N in place of M.
//   C/D (16x16): VGPR r -> M=r (lanes 0-15) / M=r+8 (lanes 16-31), N=lane&15.
// ---------------------------------------------------------------------------
__global__ __launch_bounds__(256) void sage_gemm_wmma(
        const float* __restrict__ x,
        const float* __restrict__ agg,
        const float* __restrict__ cnt,
        const float* __restrict__ Wl,
        const float* __restrict__ Wr,
        const float* __restrict__ bias,
        float*       __restrict__ out) {
    __shared__ v2f sWl2[NPAIRS];   // 32 KB
    __shared__ v2f sWr2[NPAIRS];   // 32 KB

    // Stage + swizzle weights into fragment order.
    for (int i = threadIdx.x; i < NPAIRS; i += 256) {
        const int k0g  = i >> 7;         // k0 / 4
        const int half = (i >> 6) & 1;
        const int col  = i & 63;
        const int kk   = k0g * 4 + half * 2;
        sWl2[i] = (v2f){ Wl[kk * D_OUT + col], Wl[(kk + 1) * D_OUT + col] };
        sWr2[i] = (v2f){ Wr[kk * D_OUT + col], Wr[(kk + 1) * D_OUT + col] };
    }
    __syncthreads();

    const int wave = threadIdx.x >> 5;
    const int lane = threadIdx.x & 31;
    const int tile = blockIdx.x * 8 + wave;      // uniform per wave
    if (tile >= N_TILES) return;                 // whole-wave branch, EXEC stays full

    const int m0   = tile * 16;
    const int half = lane >> 4;                  // 0: K pair (0,1), 1: K pair (2,3)
    const int lid  = lane & 15;

    const int rowA = m0 + lid;
    const float invc = 1.0f / fmaxf(cnt[rowA], 1.0f);
    const float* aggRow = agg + (size_t)rowA * D_IN;
    const float* xRow   = x   + (size_t)rowA * D_IN;

    v8f c[4];
    #pragma unroll
    for (int n = 0; n < 4; ++n) c[n] = (v8f){0.f,0.f,0.f,0.f,0.f,0.f,0.f,0.f};

    #pragma unroll 4
    for (int k0 = 0; k0 < D_IN; k0 += 4) {
        const int kk = k0 + half * 2;            // always even -> b64-aligned
        v2f aAgg = *(const v2f*)(aggRow + kk);
        v2f aX   = *(const v2f*)(xRow   + kk);
        aAgg = aAgg * invc;                      // one v_pk_mul_f32

        const int pbase = (k0 >> 2) * 128 + half * 64 + lid;
        #pragma unroll
        for (int n = 0; n < 4; ++n) {
            const v2f bl = sWl2[pbase + n * 16]; // single ds_load_b64 each
            const v2f br = sWr2[pbase + n * 16];
            c[n] = __builtin_amdgcn_wmma_f32_16x16x4_f32(
                       false, aAgg, false, bl, (short)0, c[n], false, false);
            c[n] = __builtin_amdgcn_wmma_f32_16x16x4_f32(
                       false, aX,   false, br, (short)0, c[n], false, false);
        }
    }

    // Bias + ReLU + store
    #pragma unroll
    for (int n = 0; n < 4; ++n) {
        const int col = n * 16 + lid;
        const float bv = bias[col];
        #pragma unroll
        for (int r = 0; r < 8; ++r) {
            const int row = m0 + r + half * 8;
            const float v = c[n][r] + bv;
            out[(size_t)row * D_OUT + col] = fmaxf(v, 0.0f);
        }
    }
}

// ---------------------------------------------------------------------------
extern "C" void kernel_launch(void* const* d_in, const int* in_sizes, int n_in,
                              void* d_out, int out_size, void* d_ws, size_t ws_size,
                              hipStream_t stream) {
    const float* x    = (const float*)d_in[0];
    const int*   ei   = (const int*)  d_in[1];   // [2, E] int32, row0=src row1=dst
    const float* Wl   = (const float*)d_in[2];
    const float* Wr   = (const float*)d_in[3];
    const float* bias = (const float*)d_in[4];
    float*       out  = (float*)d_out;

    const int E = in_sizes[1] / 2;
    const int* src = ei;
    const int* dst = ei + E;

    float* agg = (float*)d_ws;                       // N_NODES * D_IN
    float* cnt = agg + (size_t)N_NODES * D_IN;       // N_NODES

    // 1) zero scratch
    const int total = N_NODES * D_IN + N_NODES;
    sage_zero_ws<<<(total + 255) / 256, 256, 0, stream>>>((float*)d_ws, total);

    // 2) edge scatter (wave per edge)
    const long long threads = (long long)E * 32;
    const int blocks = (int)((threads + 255) / 256);
    sage_edge_scatter<<<blocks, 256, 0, stream>>>(x, src, dst, agg, cnt, E);

    // 3) fused normalize + dual GEMM + bias + relu via WMMA
    const int gblocks = (N_TILES + 7) / 8;
    sage_gemm_wmma<<<gblocks, 256, 0, stream>>>(x, agg, cnt, Wl, Wr, bias, out);
}